// GNN_21328807592482
// MI455X (gfx1250) — compile-verified
//
#include <hip/hip_runtime.h>
#include <hip/hip_bf16.h>
#include <math.h>

// ---------------------------------------------------------------------------
// GNN layer: scatter-mean over edges, fused Linear(3D->OUT)+tanh via f32 WMMA,
// global Frobenius normalization.  MI455X / gfx1250, wave32.
//
// out[m,n] = tanh( sum_k feats[m,k]*(W[n,k]+W[n,256+k])
//                + sum_k h[m,k]*W[n,128+k] + b[n] ) / ||out||_F
// where h = scatter_mean(feats[src] -> dst).
// ---------------------------------------------------------------------------

typedef float v2f __attribute__((ext_vector_type(2)));
typedef float v8f __attribute__((ext_vector_type(8)));

#define DIMS 128          // D == OUT == 128
#define LDS_PITCH 132     // 128 + 4 pad -> conflict-free ds_load_b64 A-fragments

__device__ __forceinline__ float atomAddF(float* p, float v) {
    return __hip_atomic_fetch_add(p, v, __ATOMIC_RELAXED, __HIP_MEMORY_SCOPE_AGENT);
}

// ---- zero-init workspace region --------------------------------------------
__global__ void gnn_zero_kernel(float* __restrict__ p, int count) {
    int i = blockIdx.x * 256 + threadIdx.x;
    if (i < count) p[i] = 0.0f;
}

// ---- fold W into two 128x128 B-matrices in WMMA-fragment-packed order ------
// Packed layout: B[(k>>1)*DIMS*2 + n*2 + (k&1)]  -> per-lane (k,k+1) pair is a
// contiguous 8B float2, so the GEMM B-fragment load is one global_load_b64.
__global__ void gnn_wprep_kernel(const float* __restrict__ W,
                                 float* __restrict__ Ba,
                                 float* __restrict__ Bh) {
    int idx = blockIdx.x * 256 + threadIdx.x;     // 128*128 threads
    int n = idx >> 7;
    int k = idx & 127;
    const float* wr = W + n * 384;
    int p = (k >> 1) * DIMS * 2 + n * 2 + (k & 1);
    Ba[p] = wr[k] + wr[256 + k];
    Bh[p] = wr[128 + k];
}

// ---- scatter-add of source rows into msg_sum + degree count ----------------
// One thread per (edge, 4-float chunk): 32 threads/edge, float4 gather,
// 4x global_atomic_add_f32 scatter (relaxed, agent scope).
__global__ __launch_bounds__(256) void gnn_scatter_kernel(
        const float* __restrict__ feats,
        const int* __restrict__ src, const int* __restrict__ dst,
        float* __restrict__ msg, float* __restrict__ deg, int nEdges) {
    int idx = blockIdx.x * 256 + threadIdx.x;
    int e  = idx >> 5;
    int ch = idx & 31;
    if (e >= nEdges) return;
    int s = src[e];
    int t = dst[e];
    float4 v = ((const float4*)(feats + (size_t)s * DIMS))[ch];
    float* mrow = msg + (size_t)t * DIMS + ch * 4;
    atomAddF(mrow + 0, v.x);
    atomAddF(mrow + 1, v.y);
    atomAddF(mrow + 2, v.z);
    atomAddF(mrow + 3, v.w);
    if (ch == 0) atomAddF(deg + t, 1.0f);
}

// ---- fused h-compute + dual GEMM (f32 WMMA) + bias + tanh + sum(y^2) -------
// Block = 256 threads = 8 wave32; block handles 16 node rows x 128 out cols.
// Wave w computes the 16x16 tile at output columns [16w, 16w+16).
__global__ __launch_bounds__(256) void gnn_gemm_kernel(
        const float* __restrict__ feats,
        const float* __restrict__ msg, const float* __restrict__ deg,
        const float* __restrict__ Ba, const float* __restrict__ Bh,
        const float* __restrict__ bias,
        float* __restrict__ out, float* __restrict__ normsq, int nNodes) {
    __shared__ float ldsX[16 * LDS_PITCH];
    __shared__ float ldsH[16 * LDS_PITCH];
    __shared__ float ssbuf[256];

    const int tid  = threadIdx.x;
    const int base = blockIdx.x * 16;

    // Stage 16x128 feats tile and h = msg/max(deg,1) tile into LDS.
    for (int i = tid; i < 16 * DIMS; i += 256) {
        int r = i >> 7;
        int d = i & 127;
        int node = base + r;
        float f = 0.0f, h = 0.0f;
        if (node < nNodes) {
            f = feats[(size_t)node * DIMS + d];
            float inv = 1.0f / fmaxf(deg[node], 1.0f);
            h = msg[(size_t)node * DIMS + d] * inv;
        }
        ldsX[r * LDS_PITCH + d] = f;
        ldsH[r * LDS_PITCH + d] = h;
    }
    __syncthreads();

    const int wave  = tid >> 5;
    const int lane  = tid & 31;
    const int m     = lane & 15;               // A-fragment row
    const int koff  = (lane >> 4) << 1;        // 0 or 2 (K split across halves)
    const int ncol  = (wave << 4) + (lane & 15);

    v8f c = {0.f, 0.f, 0.f, 0.f, 0.f, 0.f, 0.f, 0.f};

    // EXEC is all-ones here (uniform control flow) as WMMA requires.
    #pragma unroll 4
    for (int kb = 0; kb < DIMS; kb += 4) {
        const int kp = (kb + koff) >> 1;       // packed k-pair index
        // A: 16x4 f32 fragment from LDS (8B-aligned, conflict-free via pitch).
        v2f a = *(const v2f*)&ldsX[m * LDS_PITCH + kb + koff];
        v2f b = *(const v2f*)&Ba[(size_t)(kp * DIMS + ncol) * 2];
        c = __builtin_amdgcn_wmma_f32_16x16x4_f32(false, a, false, b,
                                                  (short)0, c, false, false);
        v2f a2 = *(const v2f*)&ldsH[m * LDS_PITCH + kb + koff];
        v2f b2 = *(const v2f*)&Bh[(size_t)(kp * DIMS + ncol) * 2];
        c = __builtin_amdgcn_wmma_f32_16x16x4_f32(false, a2, false, b2,
                                                  (short)0, c, false, false);
    }

    // C/D layout: c[v] is row (v + 8*(lane>=16)), col (lane&15) of the tile.
    const float bn    = bias[ncol];
    const int   mbase = (lane >> 4) << 3;      // 0 or 8
    float ss = 0.0f;
    #pragma unroll
    for (int v = 0; v < 8; ++v) {
        int row = base + mbase + v;
        float y = tanhf(c[v] + bn);
        if (row < nNodes) {
            out[(size_t)row * DIMS + ncol] = y;
            ss += y * y;
        }
    }

    // Block reduction of sum(y^2), one device atomic per block.
    ssbuf[tid] = ss;
    __syncthreads();
    for (int off = 128; off > 0; off >>= 1) {
        if (tid < off) ssbuf[tid] += ssbuf[tid + off];
        __syncthreads();
    }
    if (tid == 0) atomAddF(normsq, ssbuf[0]);
}

// ---- global normalization: out /= ||out||_F --------------------------------
__global__ void gnn_scale_kernel(float* __restrict__ out,
                                 const float* __restrict__ normsq, int count) {
    int i = blockIdx.x * 256 + threadIdx.x;
    if (i >= count) return;
    float inv = 1.0f / sqrtf(*normsq);
    out[i] *= inv;
}

// ---------------------------------------------------------------------------
extern "C" void kernel_launch(void* const* d_in, const int* in_sizes, int n_in,
                              void* d_out, int out_size, void* d_ws, size_t ws_size,
                              hipStream_t stream) {
    const float* feats = (const float*)d_in[0];   // [N, 128]
    const float* W     = (const float*)d_in[1];   // [128, 384]
    const float* bias  = (const float*)d_in[2];   // [128]
    const int*   src   = (const int*)d_in[3];     // [E]
    const int*   dst   = (const int*)d_in[4];     // [E]
    float* out = (float*)d_out;
    float* ws  = (float*)d_ws;

    const int nNodes = in_sizes[0] / DIMS;        // 100000
    const int nEdges = in_sizes[3];               // 1000000

    // Workspace layout (floats):
    //   msg    [N*128]
    //   deg    [N]
    //   normsq [1] (+3 pad for 16B alignment of Ba)
    //   Ba     [128*128]   folded W1+W3, fragment-packed
    //   Bh     [128*128]   W2, fragment-packed
    float* msg    = ws;
    float* deg    = msg + (size_t)nNodes * DIMS;
    float* normsq = deg + nNodes;
    float* Ba     = normsq + 4;
    float* Bh     = Ba + DIMS * DIMS;

    const int zeroCount = nNodes * DIMS + nNodes + 4;   // msg + deg + normsq
    gnn_zero_kernel<<<(zeroCount + 255) / 256, 256, 0, stream>>>(ws, zeroCount);

    gnn_wprep_kernel<<<(DIMS * DIMS) / 256, 256, 0, stream>>>(W, Ba, Bh);

    const long scatterThreads = (long)nEdges * 32;
    gnn_scatter_kernel<<<(scatterThreads + 255) / 256, 256, 0, stream>>>(
        feats, src, dst, msg, deg, nEdges);

    gnn_gemm_kernel<<<(nNodes + 15) / 16, 256, 0, stream>>>(
        feats, msg, deg, Ba, Bh, bias, out, normsq, nNodes);

    const int scaleCount = nNodes * DIMS;
    gnn_scale_kernel<<<(scaleCount + 255) / 256, 256, 0, stream>>>(
        out, normsq, scaleCount);
}